// LSTM_41566693490904
// MI455X (gfx1250) — compile-verified
//
#include <hip/hip_runtime.h>
#include <hip/hip_bf16.h>
#include <math.h>

typedef float v2f __attribute__((ext_vector_type(2)));
typedef float v8f __attribute__((ext_vector_type(8)));

#define B_   256
#define T_   128
#define V_   512
#define E_   64
#define H_   256
#define G4H  1024          // 4*H
#define XS   324           // LDS row stride (== 4 mod 64 -> conflict-free)
#define XS2  516           // E1 staging stride (== 4 mod 64)

__device__ __forceinline__ v8f wmma4(v2f a, v2f b, v8f c) {
  // D = A(16x4 f32) * B(4x16 f32) + C(16x16 f32)
  return __builtin_amdgcn_wmma_f32_16x16x4_f32(false, a, false, b, (short)0, c,
                                               false, false);
}
__device__ __forceinline__ v8f v8f_splat(float x) {
  v8f r;
#pragma unroll
  for (int i = 0; i < 8; ++i) r[i] = x;
  return r;
}
__device__ __forceinline__ float sigmoidf_(float x) {
  return 1.0f / (1.0f + __expf(-x));
}

// ---------------------------------------------------------------------------
// Weight pre-swizzle into WMMA B-fragment order:
//   dst[((ntile*(K/4) + k4)*32 + lane)*2 + j] = W[n*K + k]
//   n = ntile*16 + (lane&15), k = 4*k4 + 2*(lane>>4) + j    (B = W^T)
// ---------------------------------------------------------------------------
__global__ void pack_bt(const float* __restrict__ src, float* __restrict__ dst,
                        int N, int K) {
  int total = N * K;
  int K4 = K >> 2;
  for (int p = blockIdx.x * blockDim.x + threadIdx.x; p < total;
       p += gridDim.x * blockDim.x) {
    int j = p & 1, lane = (p >> 1) & 31, rest = p >> 6;
    int k4 = rest % K4, ntl = rest / K4;
    int n = (ntl << 4) + (lane & 15);
    int k = (k4 << 2) + ((lane >> 4) << 1) + j;
    dst[p] = src[n * K + k];
  }
}

// Packed [Wih_e ; Whh_e] as one K=320 operand (k<64 -> Wih, else Whh)
__global__ void pack_enc(const float* __restrict__ wih,
                         const float* __restrict__ whh,
                         float* __restrict__ dst) {
  const int total = G4H * 320;
  for (int p = blockIdx.x * blockDim.x + threadIdx.x; p < total;
       p += gridDim.x * blockDim.x) {
    int j = p & 1, lane = (p >> 1) & 31, rest = p >> 6;
    int k4 = rest % 80, ntl = rest / 80;
    int n = (ntl << 4) + (lane & 15);
    int k = (k4 << 2) + ((lane >> 4) << 1) + j;
    dst[p] = (k < 64) ? wih[n * 64 + k] : whh[n * 256 + (k - 64)];
  }
}

// WcombB[v,n] = sum_e Wih_d[n,e]*(Wd[e,v]+bd[e]) + bih_d[n] + bhh_d[n]
__global__ void wcomb_kernel(const float* __restrict__ wih_d,
                             const float* __restrict__ wd,
                             const float* __restrict__ bd,
                             const float* __restrict__ bih_d,
                             const float* __restrict__ bhh_d,
                             float* __restrict__ wcombB) {
  int idx = blockIdx.x * blockDim.x + threadIdx.x;
  if (idx >= V_ * G4H) return;
  int n = idx & 1023, v = idx >> 10;
  float s = bih_d[n] + bhh_d[n];
#pragma unroll 8
  for (int e = 0; e < 64; ++e)
    s += wih_d[n * 64 + e] * (wd[e * 512 + v] + bd[e]);
  wcombB[idx] = s;
}

// emb0[b,e] = x[b,0,:] @ Wd[e,:] + bd[e]
__global__ void emb0_kernel(const float* __restrict__ x,
                            const float* __restrict__ wd,
                            const float* __restrict__ bd,
                            float* __restrict__ emb0) {
  int idx = blockIdx.x * blockDim.x + threadIdx.x;
  if (idx >= B_ * E_) return;
  int e = idx & 63, b = idx >> 6;
  float s = bd[e];
  for (int v = 0; v < 512; ++v)
    s += x[(size_t)b * (T_ * V_) + v] * wd[e * 512 + v];
  emb0[idx] = s;
}

// g0[b,n] = emb0[b,:] @ Wih_d[n,:] + bih_d[n] + bhh_d[n]
__global__ void g0_kernel(const float* __restrict__ emb0,
                          const float* __restrict__ wih_d,
                          const float* __restrict__ bih_d,
                          const float* __restrict__ bhh_d,
                          float* __restrict__ g0) {
  int idx = blockIdx.x * blockDim.x + threadIdx.x;
  if (idx >= B_ * G4H) return;
  int n = idx & 1023, b = idx >> 10;
  float s = bih_d[n] + bhh_d[n];
#pragma unroll 8
  for (int e = 0; e < 64; ++e) s += emb0[b * 64 + e] * wih_d[n * 64 + e];
  g0[idx] = s;
}

// y[:,0,:]=0 ; x_lens -> out ; bsum = bih_e + bhh_e
__global__ void misc_kernel(const int* __restrict__ x_lens,
                            const float* __restrict__ bih_e,
                            const float* __restrict__ bhh_e,
                            float* __restrict__ out,
                            float* __restrict__ bsum) {
  int idx = blockIdx.x * blockDim.x + threadIdx.x;
  if (idx < 131072) {                       // 256*512 zeros of y[:,0,:]
    int b = idx >> 9, v = idx & 511;
    out[(size_t)b * (T_ * V_) + v] = 0.0f;
  } else if (idx < 131072 + 256) {          // x_lens passthrough
    int b = idx - 131072;
    out[(size_t)B_ * T_ * V_ + b] = (float)x_lens[b];
  } else if (idx < 131072 + 256 + 1024) {   // fused gate bias
    int i = idx - 131072 - 256;
    bsum[i] = bih_e[i] + bhh_e[i];
  }
}

// ---------------------------------------------------------------------------
// E1: x_emb = x @ We.T + be   (M=32768, K=512, N=64)  WMMA, LDS-staged A
// grid 2048 x 64 threads (2 waves, 2 n-tiles each)
// ---------------------------------------------------------------------------
__global__ __launch_bounds__(64) void emb_gemm(const float* __restrict__ x,
                                               const float* __restrict__ pk_we,
                                               const float* __restrict__ be,
                                               float* __restrict__ x_emb) {
  __shared__ __align__(16) float xs[16 * XS2];
  const int tid = threadIdx.x;
  const int w = tid >> 5, l = tid & 31, ln = l & 15, hi = l >> 4;
  const size_t mbase = (size_t)blockIdx.x * 16;

  // stage 16x512 A tile (coalesced float4)
  for (int i = tid; i < 16 * 128; i += 64) {
    int r = i >> 7, c4 = i & 127;
    float4 v = ((const float4*)x)[(mbase + r) * 128 + c4];
    *(float4*)&xs[r * XS2 + (c4 << 2)] = v;
  }
  __syncthreads();

  const int nt0 = w << 1;
  v8f acc[2];
#pragma unroll
  for (int q = 0; q < 2; ++q) acc[q] = v8f_splat(be[((nt0 + q) << 4) + ln]);

  const v2f* pb = (const v2f*)pk_we;
  for (int k4 = 0; k4 < 128; ++k4) {
    v2f a = *(const v2f*)&xs[ln * XS2 + (k4 << 2) + (hi << 1)];
#pragma unroll
    for (int q = 0; q < 2; ++q) {
      v2f b = pb[((nt0 + q) * 128 + k4) * 32 + l];
      acc[q] = wmma4(a, b, acc[q]);
    }
  }
#pragma unroll
  for (int q = 0; q < 2; ++q)
#pragma unroll
    for (int i = 0; i < 8; ++i) {
      int row = i + (hi << 3);
      x_emb[(mbase + row) * 64 + ((nt0 + q) << 4) + ln] = acc[q][i];
    }
}

// ---------------------------------------------------------------------------
// Persistent recurrent kernel: 16 blocks x 256 threads; each block owns 16
// batch rows through 128 encoder + 127 decoder steps. Wave w owns h columns
// [32w, 32w+32) (c/h in registers) and gate n-tiles {16q + 2w + {0,1}}.
// ---------------------------------------------------------------------------
__global__ __launch_bounds__(256) void lstm_recurrent(
    const float* __restrict__ x_emb, const int* __restrict__ x_lens,
    const float* __restrict__ pk_enc, const float* __restrict__ pk_whd,
    const float* __restrict__ pk_wo, const float* __restrict__ bsum_e,
    const float* __restrict__ bo, const float* __restrict__ g0,
    const float* __restrict__ wcombB, float* __restrict__ out) {
  __shared__ __align__(16) float xh[16 * XS];   // [x_emb_t(64) | h(256)] rows
  __shared__ float s_bias[G4H];
  __shared__ float s_bo[V_];
  __shared__ int s_len[16];
  __shared__ int s_arg[16];
  __shared__ float s_red[16 * 8];
  __shared__ int s_redi[16 * 8];

  const int tid = threadIdx.x;
  const int w = tid >> 5, l = tid & 31, ln = l & 15, hi = l >> 4;
  const int base = blockIdx.x * 16;
  const size_t ENC_OFF = (size_t)B_ * T_ * V_ + B_;

  for (int i = tid; i < G4H; i += 256) s_bias[i] = bsum_e[i];
  for (int i = tid; i < V_; i += 256) s_bo[i] = bo[i];
  if (tid < 16) s_len[tid] = x_lens[base + tid];
  for (int i = tid; i < 16 * XS; i += 256) xh[i] = 0.0f;
  __syncthreads();

  int nt[8];
#pragma unroll
  for (int j = 0; j < 8; ++j) nt[j] = ((j >> 1) << 4) + (w << 1) + (j & 1);

  v8f cst[2], hreg[2];
  cst[0] = v8f_splat(0.f); cst[1] = v8f_splat(0.f);
  hreg[0] = v8f_splat(0.f); hreg[1] = v8f_splat(0.f);

  // ===================== encoder: 128 steps =====================
  for (int t = 0; t < T_; ++t) {
    for (int i = tid; i < 16 * 64; i += 256) {     // stage x_emb[:,t]
      int r = i >> 6, c = i & 63;
      xh[r * XS + c] = x_emb[((size_t)(base + r) * T_ + t) * 64 + c];
    }
    __syncthreads();

    v8f acc[8];
#pragma unroll
    for (int j = 0; j < 8; ++j) acc[j] = v8f_splat(s_bias[(nt[j] << 4) + ln]);

    const v2f* pb = (const v2f*)pk_enc;
    for (int k4 = 0; k4 < 80; ++k4) {              // K = 64 + 256
      v2f a = *(const v2f*)&xh[ln * XS + (k4 << 2) + (hi << 1)];
#pragma unroll
      for (int j = 0; j < 8; ++j) {
        v2f b = pb[((nt[j] * 80 + k4) << 5) + l];
        acc[j] = wmma4(a, b, acc[j]);
      }
    }
    __syncthreads();

#pragma unroll
    for (int p = 0; p < 2; ++p)
#pragma unroll
      for (int i = 0; i < 8; ++i) {
        int row = i + (hi << 3);
        float ig = sigmoidf_(acc[0 + p][i]);
        float fg = sigmoidf_(acc[2 + p][i]);
        float gg = tanhf(acc[4 + p][i]);
        float og = sigmoidf_(acc[6 + p][i]);
        float cn = fg * cst[p][i] + ig * gg;
        float hn = og * tanhf(cn);
        bool keep = (t < s_len[row]);
        cn = keep ? cn : cst[p][i];
        hn = keep ? hn : hreg[p][i];
        cst[p][i] = cn; hreg[p][i] = hn;
        xh[row * XS + 64 + (w << 5) + (p << 4) + ln] = hn;
      }
  }
  __syncthreads();
  for (int i = tid; i < 16 * H_; i += 256) {       // encoded = h_enc
    int r = i >> 8, c = i & 255;
    out[ENC_OFF + (size_t)(base + r) * H_ + c] = xh[r * XS + 64 + c];
  }

  // ===================== decoder: 127 steps =====================
  for (int s = 0; s < T_ - 1; ++s) {
    __syncthreads();
    v8f acc[8];
    if (s == 0) {
#pragma unroll
      for (int j = 0; j < 8; ++j)
#pragma unroll
        for (int i = 0; i < 8; ++i) {
          int row = i + (hi << 3);
          acc[j][i] = g0[(size_t)(base + row) * G4H + (nt[j] << 4) + ln];
        }
    } else {
#pragma unroll
      for (int i = 0; i < 8; ++i) {
        int row = i + (hi << 3);
        int v = s_arg[row];                        // one-hot feedback gather
#pragma unroll
        for (int j = 0; j < 8; ++j)
          acc[j][i] = wcombB[(size_t)v * G4H + (nt[j] << 4) + ln];
      }
    }

    const v2f* pb = (const v2f*)pk_whd;
    for (int k4 = 0; k4 < 64; ++k4) {              // K = 256 (h only)
      v2f a = *(const v2f*)&xh[ln * XS + 64 + (k4 << 2) + (hi << 1)];
#pragma unroll
      for (int j = 0; j < 8; ++j) {
        v2f b = pb[(((nt[j] << 6) + k4) << 5) + l];
        acc[j] = wmma4(a, b, acc[j]);
      }
    }
    __syncthreads();

#pragma unroll
    for (int p = 0; p < 2; ++p)
#pragma unroll
      for (int i = 0; i < 8; ++i) {
        int row = i + (hi << 3);
        float ig = sigmoidf_(acc[0 + p][i]);
        float fg = sigmoidf_(acc[2 + p][i]);
        float gg = tanhf(acc[4 + p][i]);
        float og = sigmoidf_(acc[6 + p][i]);
        float cn = fg * cst[p][i] + ig * gg;
        float hn = og * tanhf(cn);
        cst[p][i] = cn; hreg[p][i] = hn;
        xh[row * XS + 64 + (w << 5) + (p << 4) + ln] = hn;
      }
    __syncthreads();

    // out = h @ Wo.T + bo ; wave w owns output cols [64w, 64w+64)
    v8f oacc[4];
#pragma unroll
    for (int q = 0; q < 4; ++q)
      oacc[q] = v8f_splat(s_bo[(((w << 2) + q) << 4) + ln]);
    const v2f* po = (const v2f*)pk_wo;
    for (int k4 = 0; k4 < 64; ++k4) {
      v2f a = *(const v2f*)&xh[ln * XS + 64 + (k4 << 2) + (hi << 1)];
#pragma unroll
      for (int q = 0; q < 4; ++q) {
        v2f b = po[(((((w << 2) + q) << 6) + k4) << 5) + l];
        oacc[q] = wmma4(a, b, oacc[q]);
      }
    }
    // write y[:, s+1, cols] and per-wave argmax
#pragma unroll
    for (int q = 0; q < 4; ++q)
#pragma unroll
      for (int i = 0; i < 8; ++i) {
        int row = i + (hi << 3);
        int col = (((w << 2) + q) << 4) + ln;
        out[((size_t)(base + row) * T_ + (s + 1)) * V_ + col] = oacc[q][i];
      }
#pragma unroll
    for (int i = 0; i < 8; ++i) {
      float bv = oacc[0][i];
      int bi = (w << 6) + ln;
#pragma unroll
      for (int q = 1; q < 4; ++q) {
        float v2 = oacc[q][i];
        int i2 = (w << 6) + (q << 4) + ln;
        if (v2 > bv) { bv = v2; bi = i2; }
      }
      for (int off = 8; off > 0; off >>= 1) {      // width-16 butterfly
        float ov = __shfl_xor(bv, off, 16);
        int oi = __shfl_xor(bi, off, 16);
        if (ov > bv || (ov == bv && oi < bi)) { bv = ov; bi = oi; }
      }
      if (ln == 0) {
        int row = i + (hi << 3);
        s_red[row * 8 + w] = bv;
        s_redi[row * 8 + w] = bi;
      }
    }
    __syncthreads();
    if (tid < 16) {                                // cross-wave argmax
      float bv = s_red[tid * 8];
      int bi = s_redi[tid * 8];
      for (int u = 1; u < 8; ++u) {
        float v2 = s_red[tid * 8 + u];
        int i2 = s_redi[tid * 8 + u];
        if (v2 > bv || (v2 == bv && i2 < bi)) { bv = v2; bi = i2; }
      }
      s_arg[tid] = bi;
    }
  }
}

// ---------------------------------------------------------------------------
extern "C" void kernel_launch(void* const* d_in, const int* in_sizes, int n_in,
                              void* d_out, int out_size, void* d_ws,
                              size_t ws_size, hipStream_t stream) {
  const float* x     = (const float*)d_in[0];
  const int* x_lens  = (const int*)d_in[1];
  const float* We    = (const float*)d_in[2];
  const float* be    = (const float*)d_in[3];
  const float* Wih_e = (const float*)d_in[4];
  const float* Whh_e = (const float*)d_in[5];
  const float* bih_e = (const float*)d_in[6];
  const float* bhh_e = (const float*)d_in[7];
  const float* Wd    = (const float*)d_in[8];
  const float* bd    = (const float*)d_in[9];
  const float* Wih_d = (const float*)d_in[10];
  const float* Whh_d = (const float*)d_in[11];
  const float* bih_d = (const float*)d_in[12];
  const float* bhh_d = (const float*)d_in[13];
  const float* Wo    = (const float*)d_in[14];
  const float* bo    = (const float*)d_in[15];
  float* out = (float*)d_out;

  char* ws = (char*)d_ws;
  float* ws_xemb  = (float*)(ws);                         // 32768*64
  float* ws_pkenc = (float*)(ws + 8388608);               // 1024*320
  float* ws_pkwhd = (float*)(ws + 9699328);               // 1024*256
  float* ws_pkwo  = (float*)(ws + 10747904);              // 512*256
  float* ws_pkwe  = (float*)(ws + 11272192);              // 64*512
  float* ws_wcomb = (float*)(ws + 11403264);              // 512*1024
  float* ws_emb0  = (float*)(ws + 13500416);              // 256*64
  float* ws_g0    = (float*)(ws + 13565952);              // 256*1024
  float* ws_bsum  = (float*)(ws + 14614528);              // 1024

  misc_kernel<<<(131072 + 256 + 1024 + 255) / 256, 256, 0, stream>>>(
      x_lens, bih_e, bhh_e, out, ws_bsum);
  pack_enc<<<1280, 256, 0, stream>>>(Wih_e, Whh_e, ws_pkenc);
  pack_bt<<<1024, 256, 0, stream>>>(Whh_d, ws_pkwhd, 1024, 256);
  pack_bt<<<512, 256, 0, stream>>>(Wo, ws_pkwo, 512, 256);
  pack_bt<<<128, 256, 0, stream>>>(We, ws_pkwe, 64, 512);
  wcomb_kernel<<<2048, 256, 0, stream>>>(Wih_d, Wd, bd, bih_d, bhh_d, ws_wcomb);
  emb0_kernel<<<64, 256, 0, stream>>>(x, Wd, bd, ws_emb0);
  g0_kernel<<<1024, 256, 0, stream>>>(ws_emb0, Wih_d, bih_d, bhh_d, ws_g0);
  emb_gemm<<<2048, 64, 0, stream>>>(x, ws_pkwe, be, ws_xemb);
  lstm_recurrent<<<16, 256, 0, stream>>>(ws_xemb, x_lens, ws_pkenc, ws_pkwhd,
                                         ws_pkwo, ws_bsum, bo, ws_g0, ws_wcomb,
                                         out);
}